// Offset_loss_79276506350071
// MI455X (gfx1250) — compile-verified
//
#include <hip/hip_runtime.h>
#include <stdint.h>

typedef __attribute__((ext_vector_type(2))) float v2f;
typedef __attribute__((ext_vector_type(8))) float v8f;

#define TILE    64
#define LROWS   66      // 64 output rows + 2 halo
#define LSTRIDE 72      // floats per LDS row (68 used + pad)
#define CHUNKS  17      // 68 cols / 4 floats per B128

// ---------------------------------------------------------------------------
// Peak reduction over one pyramid level. Each block handles a 64x64 output
// tile of the channel-2 heatmap plane of one batch image. The (66 x 68) input
// region is staged into LDS with CDNA5 async global->LDS B128 copies, then
// each thread walks 16 rows of one column with a rolling 3x3 window.
// acc[n*4 + {0,1,2}] accumulates {R*sum_i, R*sum_j, count} via atomics.
// ---------------------------------------------------------------------------
__global__ __launch_bounds__(256) void peak_kernel(const float* __restrict__ p,
                                                   int H, int W, float R,
                                                   float* __restrict__ acc)
{
    __shared__ float tile[LROWS * LSTRIDE];
    __shared__ float red[3];

    const int n  = blockIdx.z;
    const int r0 = blockIdx.y * TILE;
    const int c0 = blockIdx.x * TILE;
    const int t  = threadIdx.x;

    if (t < 3) red[t] = 0.0f;

    const float* plane = p + ((size_t)n * 3 + 2) * (size_t)H * (size_t)W;

    // --- async stage: rows [r0-1, r0+65), cols [c0-4, c0+64) in 16B chunks ---
    for (int task = t; task < LROWS * CHUNKS; task += 256) {
        int row   = task / CHUNKS;
        int chunk = task - row * CHUNKS;
        int gr = r0 - 1 + row;
        int gc = c0 - 4 + chunk * 4;          // 16B-aligned (W % 4 == 0)
        if (gr >= 0 && gr < H && gc >= 0) {   // gc+3 <= c0+63 <= W-1 always
            uint32_t laddr = (uint32_t)(uintptr_t)(&tile[row * LSTRIDE + chunk * 4]);
            uint64_t gaddr = (uint64_t)(uintptr_t)(plane + (size_t)gr * W + gc);
            asm volatile("global_load_async_to_lds_b128 %0, %1, off"
                         :: "v"(laddr), "v"(gaddr) : "memory");
        }
    }
    asm volatile("s_wait_asynccnt 0x0" ::: "memory");
    __syncthreads();

    // --- 3x3 strict-max stencil, rolling register window down 16 rows ---
    const int colg   = t & 63;        // column within tile
    const int rowgrp = t >> 6;        // 0..3 -> rows rowgrp*16 .. +15
    const int j      = c0 + colg;
    const int lc     = colg + 4;      // LDS column of center
    const int base   = rowgrp * 16;   // LDS row of "above" for first output row

    const bool jok = (j >= 1) && (j <= W - 2);

    float si = 0.0f, sj = 0.0f, cnt = 0.0f;

    float a0 = tile[(base    ) * LSTRIDE + lc - 1];
    float a1 = tile[(base    ) * LSTRIDE + lc    ];
    float a2 = tile[(base    ) * LSTRIDE + lc + 1];
    float b0 = tile[(base + 1) * LSTRIDE + lc - 1];
    float b1 = tile[(base + 1) * LSTRIDE + lc    ];
    float b2 = tile[(base + 1) * LSTRIDE + lc + 1];

#pragma unroll
    for (int rr = 0; rr < 16; ++rr) {
        const int lr2 = base + 2 + rr;
        float d0 = tile[lr2 * LSTRIDE + lc - 1];
        float d1 = tile[lr2 * LSTRIDE + lc    ];
        float d2 = tile[lr2 * LSTRIDE + lc + 1];

        const int  i  = r0 + base + rr;
        const bool ok = jok && (i >= 1) && (i <= H - 2);
        const float c = b1;
        const bool peak = ok &&
            (c > a0) && (c > a1) && (c > a2) &&
            (c > b0) &&             (c > b2) &&
            (c > d0) && (c > d1) && (c > d2);
        if (peak) { si += (float)i; sj += (float)j; cnt += 1.0f; }

        a0 = b0; a1 = b1; a2 = b2;
        b0 = d0; b1 = d1; b2 = d2;
    }

    // wave32 shuffle reduce, then LDS, then one global atomic triple per block
#pragma unroll
    for (int off = 16; off > 0; off >>= 1) {
        si  += __shfl_down(si,  off, 32);
        sj  += __shfl_down(sj,  off, 32);
        cnt += __shfl_down(cnt, off, 32);
    }
    if ((t & 31) == 0) {
        atomicAdd(&red[0], si);
        atomicAdd(&red[1], sj);
        atomicAdd(&red[2], cnt);
    }
    __syncthreads();
    if (t == 0) {
        atomicAdd(&acc[n * 4 + 0], R * red[0]);
        atomicAdd(&acc[n * 4 + 1], R * red[1]);
        atomicAdd(&acc[n * 4 + 2], red[2]);
    }
}

// ---------------------------------------------------------------------------
// truths_sum[n][c] = sum_box 0.5*(target[n,box,c] + target[n,box,c+2])
//                  = sum_{e=0..159} target[n,e] * w_c[e % 5]
// -> exact 64x160 x 160x2 fp32 GEMM via chained v_wmma_f32_16x16x4_f32.
// One wave per 16 batch rows (128 threads total).
//
// B operand only depends on (ncol, khalf, k0 % 5). All 5 residue cases are
// precomputed into registers before the loop, so the unrolled inner loop is
// just: global_load_b64 (A pair) -> v_wmma_f32_16x16x4_f32.
// ---------------------------------------------------------------------------
__global__ __launch_bounds__(128) void target_wmma_kernel(const float* __restrict__ tgt,
                                                          float* __restrict__ truths)
{
    const int lane  = threadIdx.x & 31;
    const int wave  = threadIdx.x >> 5;
    const int n0    = wave * 16;
    const int mrow  = n0 + (lane & 15);       // A-matrix row (batch index)
    const int khalf = (lane >> 4) & 1;        // lanes 16-31 hold K+2,K+3
    const int ncol  = lane & 15;              // B/D column

    // per-lane column weights by K-residue: w[0]=w[2]=wA, w[1]=w[3]=wB, w[4]=0
    const float wA = (ncol == 0) ? 0.5f : 0.0f;   // one v_cndmask
    const float wB = (ncol == 1) ? 0.5f : 0.0f;   // one v_cndmask
    const float w[5] = { wA, wB, wA, wB, 0.0f };

    // khalf-resolved B pair for each possible (k0 % 5); constant-indexed below
    v2f bv[5];
#pragma unroll
    for (int m = 0; m < 5; ++m) {
        bv[m].x = khalf ? w[(m + 2) % 5] : w[m];
        bv[m].y = khalf ? w[(m + 3) % 5] : w[(m + 1) % 5];
    }

    const float* arow = tgt + mrow * 160 + khalf * 2;  // per-lane A base

    v8f c = {};  // accumulator starts at zero
#pragma unroll
    for (int k0 = 0; k0 < 160; k0 += 4) {
        v2f a;
        a.x = arow[k0];                       // merged to one global_load_b64
        a.y = arow[k0 + 1];
        const v2f b = bv[k0 % 5];             // compile-time index after unroll
        c = __builtin_amdgcn_wmma_f32_16x16x4_f32(
                /*neg_a=*/false, a, /*neg_b=*/false, b,
                /*c_mod=*/(short)0, c, /*reuse_a=*/false, /*reuse_b=*/false);
    }

    // D layout: VGPR r, lanes 0-15 -> M=r, lanes 16-31 -> M=r+8; N = lane&15
    if (ncol < 2) {
        const int rbase = n0 + (khalf ? 8 : 0);
#pragma unroll
        for (int r = 0; r < 8; ++r)
            truths[(rbase + r) * 2 + ncol] = c[r];
    }
}

// ---------------------------------------------------------------------------
// Final combine: smooth-L1 sums per component, totals, sign trick, divide by
// total peak count. One block of exactly N=64 threads.
// ---------------------------------------------------------------------------
__global__ __launch_bounds__(64) void loss_kernel(const float* __restrict__ acc,
                                                  const float* __restrict__ truths,
                                                  float* __restrict__ out)
{
    __shared__ float s[7];
    const int t = threadIdx.x;
    if (t < 7) s[t] = 0.0f;
    __syncthreads();

    float cs0 = acc[t * 4 + 0];
    float cs1 = acc[t * 4 + 1];
    float cnt = acc[t * 4 + 2];
    float ts0 = truths[t * 2 + 0];
    float ts1 = truths[t * 2 + 1];

    float d0 = fabsf(cs0 - ts0);
    float d1 = fabsf(cs1 - ts1);
    float l0 = (d0 < 1.0f) ? 0.5f * d0 * d0 : d0 - 0.5f;
    float l1 = (d1 < 1.0f) ? 0.5f * d1 * d1 : d1 - 0.5f;

    atomicAdd(&s[0], l0);
    atomicAdd(&s[1], l1);
    atomicAdd(&s[2], cs0);
    atomicAdd(&s[3], cs1);
    atomicAdd(&s[4], ts0);
    atomicAdd(&s[5], ts1);
    atomicAdd(&s[6], cnt);
    __syncthreads();

    if (t == 0) {
        float offx = s[0], offy = s[1];
        float loss = (offx / fabsf(offx) * (s[2] - s[4]) +
                      offy / fabsf(offy) * (s[3] - s[5])) / s[6];
        out[0] = loss;
    }
}

// ---------------------------------------------------------------------------
extern "C" void kernel_launch(void* const* d_in, const int* in_sizes, int n_in,
                              void* d_out, int out_size, void* d_ws, size_t ws_size,
                              hipStream_t stream)
{
    const float* p1  = (const float*)d_in[0];   // [64,3,256,256]
    const float* p2  = (const float*)d_in[1];   // [64,3,128,128]
    const float* p3  = (const float*)d_in[2];   // [64,3,64,64]
    const float* tgt = (const float*)d_in[3];   // [64,32,5]

    float* acc    = (float*)d_ws;   // 64*4 floats: R*sum_i, R*sum_j, count, pad
    float* truths = acc + 256;      // 64*2 floats

    hipMemsetAsync(acc, 0, 256 * sizeof(float), stream);

    peak_kernel<<<dim3(4, 4, 64), 256, 0, stream>>>(p1, 256, 256,  4.0f, acc);
    peak_kernel<<<dim3(2, 2, 64), 256, 0, stream>>>(p2, 128, 128,  8.0f, acc);
    peak_kernel<<<dim3(1, 1, 64), 256, 0, stream>>>(p3,  64,  64, 16.0f, acc);

    target_wmma_kernel<<<1, 128, 0, stream>>>(tgt, truths);
    loss_kernel<<<1, 64, 0, stream>>>(acc, truths, (float*)d_out);
}